// SparseGCM_31078383354018
// MI455X (gfx1250) — compile-verified
//
#include <hip/hip_runtime.h>

// ---------------- problem constants ----------------
#define Bdim  32
#define Tt    64
#define FEATc 256
#define Nn    2048
#define Ee    262144

// d_out layout (flat 4-byte element offsets), in reference return order:
// mx_dense (B*t*FEAT) | nodes (B*N*FEAT) | edges (2*E int bits) | weights (E) | T+taus (B int bits)
#define MX_OFF     0
#define NODES_OFF  (Bdim * Tt * FEATc)                  //   524288
#define EDGES_OFF  (NODES_OFF + Bdim * Nn * FEATc)      // 17301504
#define WTS_OFF    (EDGES_OFF + 2 * Ee)                 // 17825792
#define TT_OFF     (WTS_OFF + Ee)                       // 18087936

// ws layout (4-byte elements): agg[B*t*FEAT] | counter | pad | edge list (3 ints per entry)
#define AGG_ELEMS  (Bdim * Tt * FEATc)                  // 524288 floats
#define CNT_IDX    AGG_ELEMS
#define LIST_IDX   (AGG_ELEMS + 4)                      // 16B aligned

typedef __attribute__((ext_vector_type(2))) float v2f;
typedef __attribute__((ext_vector_type(8))) float v8f;

// ---------------- 1: passthrough copy of nodes (dominant traffic, float4) ----------------
__global__ void sgcm_copy_nodes(const float4* __restrict__ src, float4* __restrict__ dst, int n4) {
    int i = blockIdx.x * blockDim.x + threadIdx.x;
    int stride = gridDim.x * blockDim.x;
    for (; i < n4; i += stride) dst[i] = src[i];
}

// ---------------- 2: scatter x rows into updated nodes ----------------
// grid = B*t blocks, 64 threads (one float4 each)
__global__ void sgcm_scatter_x(const float4* __restrict__ x, const int* __restrict__ taus,
                               const int* __restrict__ T, float4* __restrict__ out_nodes) {
    int blk = blockIdx.x;
    int b = blk >> 6, j = blk & 63;
    if (j >= taus[b]) return;
    size_t g = (size_t)b * Nn + T[b] + j;
    out_nodes[g * (FEATc / 4) + threadIdx.x] =
        x[((size_t)b * Tt + j) * (FEATc / 4) + threadIdx.x];
}

// ---------------- 3: zero agg + counter ----------------
__global__ void sgcm_zero_ws(float* __restrict__ ws, int n) {
    int i = blockIdx.x * blockDim.x + threadIdx.x;
    int stride = gridDim.x * blockDim.x;
    for (; i < n; i += stride) ws[i] = 0.0f;
}

// ---------------- 4: filter edges whose dst hits an active window row ----------------
__global__ void sgcm_filter_edges(const int* __restrict__ edges, const int* __restrict__ T,
                                  const int* __restrict__ taus, const float* __restrict__ wts,
                                  int* __restrict__ cnt, int* __restrict__ list, int cap) {
    int e = blockIdx.x * blockDim.x + threadIdx.x;
    int stride = gridDim.x * blockDim.x;
    for (; e < Ee; e += stride) {
        int d   = edges[Ee + e];          // dst
        int b   = d >> 11;                // / N
        int rel = (d & (Nn - 1)) - T[b];
        if (rel >= 0 && rel < taus[b]) {  // only unmasked rows matter
            int pos = atomicAdd(cnt, 1);
            if (pos < cap) {
                list[3 * pos + 0] = edges[e];           // src node row
                list[3 * pos + 1] = b * Tt + rel;       // compact agg row
                list[3 * pos + 2] = __float_as_int(wts[e]);
            }
        }
    }
}

// ---------------- 5: one wave per matching edge: gather, scale, atomic-add ----------------
__global__ void sgcm_spread(const float* __restrict__ nodes_out, const int* __restrict__ list,
                            const int* __restrict__ cnt, int cap, float* __restrict__ agg) {
    int tid = blockIdx.x * blockDim.x + threadIdx.x;
    int wg = tid >> 5, lane = tid & 31;
    int nw = (gridDim.x * blockDim.x) >> 5;
    int n = *cnt; if (n > cap) n = cap;
    for (int i = wg; i < n; i += nw) {
        int   s = list[3 * i + 0];
        int   r = list[3 * i + 1];
        float w = __int_as_float(list[3 * i + 2]);
        const float4* srow = (const float4*)(nodes_out + (size_t)s * FEATc) + lane * 2;
        float4 v0 = srow[0], v1 = srow[1];
        float* arow = agg + (size_t)r * FEATc + lane * 8;
        unsafeAtomicAdd(arow + 0, v0.x * w); unsafeAtomicAdd(arow + 1, v0.y * w);
        unsafeAtomicAdd(arow + 2, v0.z * w); unsafeAtomicAdd(arow + 3, v0.w * w);
        unsafeAtomicAdd(arow + 4, v1.x * w); unsafeAtomicAdd(arow + 5, v1.y * w);
        unsafeAtomicAdd(arow + 6, v1.z * w); unsafeAtomicAdd(arow + 7, v1.w * w);
    }
}

// ---------------- 6: GEMM (2048x256)@(256x256) via V_WMMA_F32_16X16X4_F32 + tanh + mask ----
// grid = 128 blocks (one 16-row M tile), 512 threads = 16 waves (one 16-col N tile each).
#define AS_STRIDE 260  // pad: banks = (4*l + kb) -> conflict-free for both lane halves
__global__ __launch_bounds__(512) void sgcm_gemm(const float* __restrict__ nodes_out,
                                                 const float* __restrict__ agg,
                                                 const float* __restrict__ Wm,
                                                 const int* __restrict__ T,
                                                 const int* __restrict__ taus,
                                                 float* __restrict__ mx) {
    __shared__ float As[16 * AS_STRIDE];
    const int tid   = threadIdx.x;
    const int mt    = blockIdx.x;
    const int b     = mt >> 2;               // 4 M-tiles per batch (64 rows / 16)
    const int jbase = (mt & 3) << 4;
    const int Tb    = T[b];
    const int tau   = taus[b];

    // stage A tile: h = flat[row] + agg[row], 16 rows x 256 K
    {
        int r = tid >> 5;            // 0..15
        int c = (tid & 31) * 8;      // 0..248
        int j = jbase + r;
        const float* frow = nodes_out + ((size_t)b * Nn + Tb + j) * FEATc + c;
        const float* arow = agg + ((size_t)b * Tt + j) * FEATc + c;
        float* dstp = &As[r * AS_STRIDE + c];
#pragma unroll
        for (int q = 0; q < 8; q += 4) {
            float4 fv = *(const float4*)(frow + q);
            float4 av = *(const float4*)(arow + q);
            dstp[q + 0] = fv.x + av.x; dstp[q + 1] = fv.y + av.y;
            dstp[q + 2] = fv.z + av.z; dstp[q + 3] = fv.w + av.w;
        }
    }
    __syncthreads();

    const int nt   = tid >> 5;   // wave id == N tile
    const int lane = tid & 31;
    const int l    = lane & 15;
    const int half = lane >> 4;
    const float* Wcol = Wm + nt * 16 + l;    // W[k][n], row-major, col n fixed

    v8f acc = {0.f, 0.f, 0.f, 0.f, 0.f, 0.f, 0.f, 0.f};
#pragma unroll 4
    for (int k0 = 0; k0 < FEATc; k0 += 4) {
        const int kb = k0 + half * 2;        // ISA 16x4 A layout: VGPR0/1 = K {kb, kb+1}
        v2f a;  a.x  = As[l * AS_STRIDE + kb];      a.y  = As[l * AS_STRIDE + kb + 1];
        v2f bv; bv.x = Wcol[(size_t)kb * FEATc];    bv.y = Wcol[(size_t)(kb + 1) * FEATc];
        acc = __builtin_amdgcn_wmma_f32_16x16x4_f32(false, a, false, bv,
                                                    (short)0, acc, false, false);
    }

    // D layout: VGPR v, lanes 0-15 -> M=v, lanes 16-31 -> M=v+8; N = l
    float* outp = mx + (size_t)b * Tt * FEATc + nt * 16 + l;
#pragma unroll
    for (int v = 0; v < 8; ++v) {
        int j = jbase + v + half * 8;
        float val = (j < tau) ? tanhf(acc[v]) : 0.0f;
        outp[(size_t)j * FEATc] = val;
    }
}

// ---------------- 7: T + taus (int bits into the f32 output buffer) ----------------
__global__ void sgcm_tt(const int* __restrict__ T, const int* __restrict__ taus,
                        int* __restrict__ out_tt) {
    int i = threadIdx.x;
    if (i < Bdim) out_tt[i] = T[i] + taus[i];
}

extern "C" void kernel_launch(void* const* d_in, const int* in_sizes, int n_in,
                              void* d_out, int out_size, void* d_ws, size_t ws_size,
                              hipStream_t stream) {
    const float* x     = (const float*)d_in[0];
    const float* nodes = (const float*)d_in[1];
    const float* wts   = (const float*)d_in[2];
    const float* Wm    = (const float*)d_in[3];
    const int*   taus  = (const int*)d_in[4];
    const int*   edges = (const int*)d_in[5];
    const int*   T     = (const int*)d_in[6];

    float* out       = (float*)d_out;
    float* out_mx    = out + MX_OFF;
    float* out_nodes = out + NODES_OFF;

    float* agg = (float*)d_ws;
    int*   cnt = (int*)d_ws + CNT_IDX;
    int*   lst = (int*)d_ws + LIST_IDX;
    long long cap_ll = ((long long)(ws_size / 4) - LIST_IDX) / 3;
    int cap = cap_ll < 0 ? 0 : (cap_ll > Ee ? Ee : (int)cap_ll);

    // passthrough outputs
    hipMemcpyAsync(out + EDGES_OFF, d_in[5], (size_t)2 * Ee * 4, hipMemcpyDeviceToDevice, stream);
    hipMemcpyAsync(out + WTS_OFF,   d_in[2], (size_t)Ee * 4,     hipMemcpyDeviceToDevice, stream);

    // 1) nodes -> out_nodes (128 MB, the roofline term)
    sgcm_copy_nodes<<<4096, 256, 0, stream>>>((const float4*)nodes, (float4*)out_nodes,
                                              Bdim * Nn * (FEATc / 4));
    // 2) overwrite scattered rows with x
    sgcm_scatter_x<<<Bdim * Tt, 64, 0, stream>>>((const float4*)x, taus, T, (float4*)out_nodes);
    // 3) zero agg + counter
    sgcm_zero_ws<<<512, 256, 0, stream>>>((float*)d_ws, LIST_IDX);
    // 4) compact the ~E*2048/65536 edges that hit active rows
    sgcm_filter_edges<<<1024, 256, 0, stream>>>(edges, T, taus, wts, cnt, lst, cap);
    // 5) gather/scale/atomic-add into compact agg
    sgcm_spread<<<1024, 256, 0, stream>>>(out_nodes, lst, cnt, cap, agg);
    // 6) WMMA GEMM + tanh + mask -> mx_dense
    sgcm_gemm<<<Bdim * Tt / 16, 512, 0, stream>>>(out_nodes, agg, Wm, T, taus, out_mx);
    // 7) T + taus
    sgcm_tt<<<1, 32, 0, stream>>>(T, taus, (int*)out + TT_OFF);
}